// GCNProtein_3384434230050
// MI455X (gfx1250) — compile-verified
//
#include <hip/hip_runtime.h>
#include <hip/hip_bf16.h>

typedef __attribute__((ext_vector_type(2))) float v2f;
typedef __attribute__((ext_vector_type(8))) float v8f;
typedef __attribute__((ext_vector_type(4))) int   v4i;

// ---------------------------------------------------------------------------
// Kernel A: pre1[i] = feat[i] * norm[i];  agg1[i] = 0
// ---------------------------------------------------------------------------
__global__ void gcn_premul_zero(const float* __restrict__ feat,
                                const float* __restrict__ norm,
                                float* __restrict__ pre1,
                                float* __restrict__ agg1,
                                int n) {
  int i = blockIdx.x * blockDim.x + threadIdx.x;
  if (i < n) {
    pre1[i] = feat[i] * norm[i];
    agg1[i] = 0.0f;
  }
}

// ---------------------------------------------------------------------------
// Edge scatter: agg[dst[e]] += val[src[e]]   (scalar feature per edge)
// 4 edges per thread via 128-bit non-temporal streaming loads of the edge
// lists (clang ext-vector type so the NT builtin accepts it).
// val[] is L2-resident (4 MB), so gathers hit L2; atomics are native f32
// no-return global_atomic_add_f32 at agent scope.
// ---------------------------------------------------------------------------
__global__ void gcn_edge_scatter(const int* __restrict__ src,
                                 const int* __restrict__ dst,
                                 const float* __restrict__ val,
                                 float* __restrict__ agg,
                                 int e4, int e) {
  int tid = blockIdx.x * blockDim.x + threadIdx.x;
  if (tid < e4) {
    const v4i* s4 = (const v4i*)src;
    const v4i* d4 = (const v4i*)dst;
    v4i s = __builtin_nontemporal_load(&s4[tid]);
    v4i d = __builtin_nontemporal_load(&d4[tid]);
    __hip_atomic_fetch_add(&agg[d.x], val[s.x], __ATOMIC_RELAXED, __HIP_MEMORY_SCOPE_AGENT);
    __hip_atomic_fetch_add(&agg[d.y], val[s.y], __ATOMIC_RELAXED, __HIP_MEMORY_SCOPE_AGENT);
    __hip_atomic_fetch_add(&agg[d.z], val[s.z], __ATOMIC_RELAXED, __HIP_MEMORY_SCOPE_AGENT);
    __hip_atomic_fetch_add(&agg[d.w], val[s.w], __ATOMIC_RELAXED, __HIP_MEMORY_SCOPE_AGENT);
  } else if (tid == e4) {
    // scalar tail (none for E = 16,000,000, kept for robustness)
    for (int k = e4 * 4; k < e; ++k) {
      __hip_atomic_fetch_add(&agg[dst[k]], val[src[k]], __ATOMIC_RELAXED, __HIP_MEMORY_SCOPE_AGENT);
    }
  }
}

// ---------------------------------------------------------------------------
// Kernel C: layer-1 epilogue with V_WMMA_F32_16X16X4_F32, 16 nodes per wave.
//   x[m]    = agg1[m] * norm[m]
//   h1[m,n] = relu(x[m]*W1[n] + b1[n]),  n = 0..2
//   t[m]    = norm[m] * sum_n h1[m,n]*W2[n]
// Also zeroes agg2 for the next layer.
//
// A (16x4 f32, documented layout): lanes 0-15 {VGPR0=K0, VGPR1=K1},
// lanes 16-31 {K2, K3}.  We place x in K0 and zero K1..K3.
// B: replicate the W1 row into every K row (robust to B's K-row lane
// placement since A's K>0 entries are zero).  C carries the bias.
// D[m][n] = x[m]*W1[n] + b1[n]; columns n>=3 are exactly 0 after relu*W2sel.
// ---------------------------------------------------------------------------
__global__ void gcn_layer1_wmma(const float* __restrict__ agg1,
                                const float* __restrict__ norm,
                                const float* __restrict__ W1,
                                const float* __restrict__ b1,
                                const float* __restrict__ W2,
                                float* __restrict__ t,
                                float* __restrict__ agg2,
                                int n_nodes) {
  const int lane = threadIdx.x & 31;
  const int wave = threadIdx.x >> 5;
  const int base = (blockIdx.x * (blockDim.x >> 5) + wave) * 16;
  const int col = lane & 15;  // N index within the 16x16 tile

  float w1n = 0.0f, b1n = 0.0f, w2n = 0.0f;
  if (col < 3) {
    w1n = W1[col];
    b1n = b1[col];
    w2n = W2[col];
  }

  // Load x for the 16 nodes of this wave (lanes 0-15 carry M=0..15, K=0).
  const int idx = base + col;
  const bool rowlane = (lane < 16) && (idx < n_nodes);
  float xv = 0.0f;
  if (rowlane) {
    xv = agg1[idx] * norm[idx];
    agg2[idx] = 0.0f;  // zero next-layer accumulator
  }

  v2f a;
  a.x = xv;      // K0 (lanes 0-15); lanes 16-31 (K2) are 0 via rowlane
  a.y = 0.0f;    // K1 / K3
  v2f b;
  b.x = w1n;     // same W1 row in every K slot
  b.y = w1n;
  v8f c;
#pragma unroll
  for (int r = 0; r < 8; ++r) c[r] = b1n;

  v8f d = __builtin_amdgcn_wmma_f32_16x16x4_f32(
      /*neg_a=*/false, a, /*neg_b=*/false, b,
      /*c_mod=*/(short)0, c, /*reuse_a=*/false, /*reuse_b=*/false);

  // D layout: VGPR r -> rows M=r (lanes 0-15) and M=r+8 (lanes 16-31),
  // N = lane&15.  relu, weight by W2[n], reduce n=0..3 with a 4-lane
  // butterfly (n=3 contributes 0), scale by norm, store one t per node.
#pragma unroll
  for (int r = 0; r < 8; ++r) {
    float h = fmaxf(d[r], 0.0f) * w2n;
    h += __shfl_xor(h, 1);
    h += __shfl_xor(h, 2);
    if (col == 0) {
      const int id = base + r + ((lane >> 4) << 3);
      if (id < n_nodes) t[id] = norm[id] * h;
    }
  }
}

// ---------------------------------------------------------------------------
// Kernel E: out[i] = relu(norm[i]*agg2[i] + b2)
// ---------------------------------------------------------------------------
__global__ void gcn_final(const float* __restrict__ norm,
                          const float* __restrict__ agg2,
                          const float* __restrict__ b2,
                          float* __restrict__ out,
                          int n) {
  int i = blockIdx.x * blockDim.x + threadIdx.x;
  if (i < n) {
    out[i] = fmaxf(norm[i] * agg2[i] + b2[0], 0.0f);
  }
}

// ---------------------------------------------------------------------------
// Launch: inputs are {feat, norm, src, dst, W1, b1, W2, b2}.
// ---------------------------------------------------------------------------
extern "C" void kernel_launch(void* const* d_in, const int* in_sizes, int n_in,
                              void* d_out, int out_size, void* d_ws, size_t ws_size,
                              hipStream_t stream) {
  (void)n_in; (void)out_size; (void)ws_size;

  const float* feat = (const float*)d_in[0];
  const float* norm = (const float*)d_in[1];
  const int*   src  = (const int*)d_in[2];
  const int*   dst  = (const int*)d_in[3];
  const float* W1   = (const float*)d_in[4];
  const float* b1   = (const float*)d_in[5];
  const float* W2   = (const float*)d_in[6];
  const float* b2   = (const float*)d_in[7];
  float* out = (float*)d_out;

  const int N = in_sizes[0];
  const int E = in_sizes[2];

  // Workspace layout: pre1 | agg1 | t | agg2  (N floats each)
  float* pre1 = (float*)d_ws;
  float* agg1 = pre1 + N;
  float* tval = agg1 + N;
  float* agg2 = tval + N;

  const int TB = 256;
  const int grid_nodes = (N + TB - 1) / TB;

  const int e4 = E / 4;
  const int edge_threads = e4 + 1;  // +1 thread covers any scalar tail
  const int grid_edges = (edge_threads + TB - 1) / TB;

  // WMMA kernel: 16 nodes per wave, 8 waves per 256-thread block -> 128 nodes/block
  const int nodes_per_block = (TB / 32) * 16;
  const int grid_wmma = (N + nodes_per_block - 1) / nodes_per_block;

  // 1) pre1 = feat*norm, agg1 = 0
  gcn_premul_zero<<<grid_nodes, TB, 0, stream>>>(feat, norm, pre1, agg1, N);
  // 2) agg1[dst] += pre1[src]
  gcn_edge_scatter<<<grid_edges, TB, 0, stream>>>(src, dst, pre1, agg1, e4, E);
  // 3) t = norm * (relu((agg1*norm)@W1 + b1) @ W2), agg2 = 0   [WMMA]
  gcn_layer1_wmma<<<grid_wmma, TB, 0, stream>>>(agg1, norm, W1, b1, W2, tval, agg2, N);
  // 4) agg2[dst] += t[src]
  gcn_edge_scatter<<<grid_edges, TB, 0, stream>>>(src, dst, tval, agg2, e4, E);
  // 5) out = relu(norm*agg2 + b2)
  gcn_final<<<grid_nodes, TB, 0, stream>>>(norm, agg2, b2, out, N);
}